// RoutingIRFAggregator_43173011259883
// MI455X (gfx1250) — compile-verified
//
#include <hip/hip_runtime.h>
#include <math.h>

// ---------------------------------------------------------------------------
// RoutingIRFAggregator for MI455X (gfx1250, wave32)
//
//   N = 100000 nodes, P = 1e6 paths, E = 8e6 entries, T = 16, F = 9
//
//   Phase A (node_kernel):  log(rfft(exp(-t/k)/k))  -> 7.2 MB table in d_ws
//   Phase B (paths_kernel): sorted-segment sum via per-path binary search,
//                           complex exp, then inverse-rFFT done as a GEMM
//                           [256 x 20] x [20 x 16] using V_WMMA_F32_16X16X4_F32
//                           (fp32 WMMA => same precision as scalar fp32 code),
//                           then relu / flip / normalize.
// ---------------------------------------------------------------------------

typedef __attribute__((ext_vector_type(2))) float v2f;
typedef __attribute__((ext_vector_type(8))) float v8f;

#define TT 16          // time taps
#define FF 9           // rfft bins
#define KPAD 20        // 18 (9 re + 9 im) padded to multiple of 4
#define BLOCK 256      // paths per block (8 waves x 2 tiles of 16)
#define PI_F 3.14159265358979323846f

// ---------------------------------------------------------------------------
// Kernel 1: per-node log spectrum table.
// logtab[n*18 + f]     = Re(log(rfft(irf_n)[f]))  (f = 0..8)
// logtab[n*18 + 9 + f] = Im(log(rfft(irf_n)[f]))
// ---------------------------------------------------------------------------
__global__ void node_logspec_kernel(const float* __restrict__ params,
                                    float* __restrict__ logtab, int N) {
    int n = blockIdx.x * blockDim.x + threadIdx.x;
    if (n >= N) return;

    float k    = params[n] + 0.5f;
    float invk = 1.0f / k;

    float irf[TT];
#pragma unroll
    for (int t = 0; t < TT; ++t)
        irf[t] = expf(-(float)t * invk) * invk;

    // twiddles: c[j] = cos(pi*j/8), s[j] = sin(pi*j/8), index (f*t) & 15
    float c[TT], s[TT];
#pragma unroll
    for (int j = 0; j < TT; ++j) {
        float a = (float)j * (PI_F / 8.0f);
        c[j] = cosf(a);
        s[j] = sinf(a);
    }

    float* dst = logtab + (size_t)n * (2 * FF);
#pragma unroll
    for (int f = 0; f < FF; ++f) {
        float re = 0.0f, im = 0.0f;
#pragma unroll
        for (int t = 0; t < TT; ++t) {
            int j = (f * t) & 15;
            re = fmaf(irf[t], c[j], re);
            im = fmaf(irf[t], -s[j], im);   // exp(-i 2pi f t / 16)
        }
        dst[f]      = 0.5f * logf(re * re + im * im);  // log|z|
        dst[FF + f] = atan2f(im, re);                  // arg(z)
    }
}

// inverse-rFFT matrix entry B[k][t], K-padded to 20.
// rows 0..8  : real-part coefficients, rows 9..17 : imag-part coefficients.
__device__ __forceinline__ float irdft_coef(int kk, int t) {
    const float inv16 = 1.0f / 16.0f;
    if (kk == 0)  return inv16;                                   // R0
    if (kk < 8)   return 2.0f * inv16 * cosf((float)(kk * t) * (PI_F / 8.0f));
    if (kk == 8)  return ((t & 1) ? -inv16 : inv16);              // R8 * (-1)^t
    if (kk >= 9 && kk < 17) {                                     // imag f=1..7
        int f = kk - 9;
        if (f == 0) return 0.0f;
        return -2.0f * inv16 * sinf((float)(f * t) * (PI_F / 8.0f));
    }
    return 0.0f;                                                  // I8, padding
}

// ---------------------------------------------------------------------------
// Kernel 2: fused segment-sum + complex exp + WMMA irFFT + relu/flip/norm.
// ---------------------------------------------------------------------------
__global__ void __launch_bounds__(BLOCK)
paths_fused_kernel(const int* __restrict__ path_idxs,
                   const int* __restrict__ path_nodes,
                   const float* __restrict__ logtab,
                   float* __restrict__ out, int E, int P) {
    __shared__ float Abuf[BLOCK][KPAD];   // freq-domain rows (WMMA A staging)
    __shared__ float Dbuf[BLOCK][17];     // time-domain rows (WMMA D staging)

    const int tid = threadIdx.x;
    const int p   = blockIdx.x * BLOCK + tid;

    // ---- segment sum of complex logs (path_idxs sorted => binary search) ----
    float acc[2 * FF];
#pragma unroll
    for (int i = 0; i < 2 * FF; ++i) acc[i] = 0.0f;

    if (p < P) {
        int lo = 0, hi = E;
        while (lo < hi) {                       // lower_bound(p)
            int mid = (lo + hi) >> 1;
            if (path_idxs[mid] < p) lo = mid + 1; else hi = mid;
        }
        int e = lo;
        hi = E;
        while (lo < hi) {                       // lower_bound(p+1)
            int mid = (lo + hi) >> 1;
            if (path_idxs[mid] < p + 1) lo = mid + 1; else hi = mid;
        }
        const int end = lo;
        for (; e < end; ++e) {                  // avg 8 entries, table is L2-hot
            const float* src = logtab + (size_t)path_nodes[e] * (2 * FF);
#pragma unroll
            for (int i = 0; i < 2 * FF; ++i) acc[i] += src[i];
        }
    }

    // ---- complex exp: exp(re + i*im) = e^re (cos im + i sin im) ----
#pragma unroll
    for (int f = 0; f < FF; ++f) {
        float mag = expf(acc[f]);
        Abuf[tid][f]      = mag * cosf(acc[FF + f]);
        Abuf[tid][FF + f] = mag * sinf(acc[FF + f]);
    }
    Abuf[tid][18] = 0.0f;
    Abuf[tid][19] = 0.0f;
    __syncthreads();

    // ---- inverse rFFT as fp32 WMMA: [16 x 20] x [20 x 16] per tile ----
    const int lane  = tid & 31;
    const int wave  = tid >> 5;
    const int mrow  = lane & 15;            // A: lanes 0-15 carry M, mirrored
    const int khalf = (lane >> 4) << 1;     // VGPR pair holds K+0/K+1 or K+2/K+3
    const int tcol  = lane & 15;            // B/D: lanes carry N (time sample)

    v2f bfrag[KPAD / 4];
#pragma unroll
    for (int cch = 0; cch < KPAD / 4; ++cch) {
        int k0 = cch * 4 + khalf;
        bfrag[cch].x = irdft_coef(k0,     tcol);
        bfrag[cch].y = irdft_coef(k0 + 1, tcol);
    }

#pragma unroll
    for (int sub = 0; sub < 2; ++sub) {     // each wave: 2 tiles of 16 paths
        const int m0 = wave * 32 + sub * 16;
        v8f d = {};
#pragma unroll
        for (int cch = 0; cch < KPAD / 4; ++cch) {
            int k0 = cch * 4 + khalf;
            v2f a;
            a.x = Abuf[m0 + mrow][k0];
            a.y = Abuf[m0 + mrow][k0 + 1];
            // D = A x B + C   (emits v_wmma_f32_16x16x4_f32)
            d = __builtin_amdgcn_wmma_f32_16x16x4_f32(
                    false, a, false, bfrag[cch], (short)0, d, false, false);
        }
        // C/D layout: VGPR r / lane<16  -> (M=r,   N=lane)
        //             VGPR r / lane>=16 -> (M=r+8, N=lane-16)
        const int mbase = m0 + ((lane >> 4) ? 8 : 0);
#pragma unroll
        for (int r = 0; r < 8; ++r)
            Dbuf[mbase + r][tcol] = d[r];
    }
    __syncthreads();

    // ---- relu, normalize, flip, store ----
    if (p < P) {
        float v[TT];
        float ssum = 0.0f;
#pragma unroll
        for (int t = 0; t < TT; ++t) {
            v[t] = fmaxf(Dbuf[tid][t], 0.0f);
            ssum += v[t];
        }
        const float inv = 1.0f / ssum;
        float4* o = (float4*)(out + (size_t)p * TT);
        o[0] = make_float4(v[15] * inv, v[14] * inv, v[13] * inv, v[12] * inv);
        o[1] = make_float4(v[11] * inv, v[10] * inv, v[9]  * inv, v[8]  * inv);
        o[2] = make_float4(v[7]  * inv, v[6]  * inv, v[5]  * inv, v[4]  * inv);
        o[3] = make_float4(v[3]  * inv, v[2]  * inv, v[1]  * inv, v[0]  * inv);
    }
}

extern "C" void kernel_launch(void* const* d_in, const int* in_sizes, int n_in,
                              void* d_out, int out_size, void* d_ws, size_t ws_size,
                              hipStream_t stream) {
    const float* params     = (const float*)d_in[0];
    const int*   path_idxs  = (const int*)d_in[1];
    const int*   path_nodes = (const int*)d_in[2];
    float*       out        = (float*)d_out;
    float*       logtab     = (float*)d_ws;     // N * 18 floats = 7.2 MB

    const int N = in_sizes[0];
    const int E = in_sizes[1];
    const int P = out_size / TT;

    node_logspec_kernel<<<(N + 255) / 256, 256, 0, stream>>>(params, logtab, N);
    paths_fused_kernel<<<(P + BLOCK - 1) / BLOCK, BLOCK, 0, stream>>>(
        path_idxs, path_nodes, logtab, out, E, P);
}